// LightGCNEncoder_11836929868661
// MI455X (gfx1250) — compile-verified
//
#include <hip/hip_runtime.h>
#include <hip/hip_bf16.h>

#define THREADS 256

typedef __attribute__((ext_vector_type(2))) float v2f;
typedef __attribute__((ext_vector_type(8))) float v8f;

__device__ __forceinline__ void atomic_add_f32(float* p, float v) {
  // Agent-scope relaxed fp32 add -> global_atomic_add_f32 on gfx1250
  __hip_atomic_fetch_add(p, v, __ATOMIC_RELAXED, __HIP_MEMORY_SCOPE_AGENT);
}

// total = x ; hA = 0 ; deg = 0
__global__ __launch_bounds__(THREADS) void lgcn_init(
    const float* __restrict__ x, float* __restrict__ total,
    float* __restrict__ hA, float* __restrict__ deg, int n128, int n) {
  int i = blockIdx.x * THREADS + threadIdx.x;
  if (i < n128) { total[i] = x[i]; hA[i] = 0.0f; }
  if (i < n) deg[i] = 0.0f;
}

// deg[dst] += 1 (in-degree at target, matching PyG gcn_norm w/o self-loops)
__global__ __launch_bounds__(THREADS) void lgcn_deg(
    const long long* __restrict__ edge, float* __restrict__ deg, int E) {
  int e = blockIdx.x * THREADS + threadIdx.x;
  if (e >= E) return;
  int d = (int)edge[(size_t)E + e];
  atomic_add_f32(&deg[d], 1.0f);
}

// in-place: deg -> deg^{-1/2} (0 where deg==0)
__global__ __launch_bounds__(THREADS) void lgcn_dis(float* __restrict__ deg, int n) {
  int i = blockIdx.x * THREADS + threadIdx.x;
  if (i >= n) return;
  float d = deg[i];
  deg[i] = (d > 0.0f) ? rsqrtf(d) : 0.0f;
}

// norm[e] = dis[src] * dis[dst]
__global__ __launch_bounds__(THREADS) void lgcn_norm(
    const long long* __restrict__ edge, const float* __restrict__ dis,
    float* __restrict__ normv, int E) {
  int e = blockIdx.x * THREADS + threadIdx.x;
  if (e >= E) return;
  int s = (int)edge[e];
  int d = (int)edge[(size_t)E + e];
  normv[e] = dis[s] * dis[d];
}

// one wave32 per edge; lane handles float4 -> coalesced 512B gather + 4 fp32 atomics
__global__ __launch_bounds__(THREADS) void lgcn_scatter(
    const long long* __restrict__ edge, const float* __restrict__ normv,
    const float* __restrict__ h_in, float* __restrict__ h_out, int E) {
  int t = blockIdx.x * THREADS + threadIdx.x;
  int e = t >> 5;
  if (e >= E) return;
  int lane = t & 31;
  int s = (int)edge[e];
  int d = (int)edge[(size_t)E + e];
  float w = normv[e];
  const float4* srcp = (const float4*)(h_in + (size_t)s * 128);
  float4 v = srcp[lane];
  float* dstp = h_out + (size_t)d * 128 + lane * 4;
  atomic_add_f32(dstp + 0, v.x * w);
  atomic_add_f32(dstp + 1, v.y * w);
  atomic_add_f32(dstp + 2, v.z * w);
  atomic_add_f32(dstp + 3, v.w * w);
}

// total += h_cur ; zero the other ping-pong buffer for the next layer
__global__ __launch_bounds__(THREADS) void lgcn_accum(
    float* __restrict__ total, const float* __restrict__ h_cur,
    float* __restrict__ h_other, int n128) {
  int i = blockIdx.x * THREADS + threadIdx.x;
  if (i >= n128) return;
  total[i] += h_cur[i];
  h_other[i] = 0.0f;
}

// out[n x 64] = (total/4) @ W[128x64] + b, via V_WMMA_F32_16X16X4_F32.
// One wave per 16-row strip, 4 column tiles (4 accumulators).
// W is staged in LDS K-pair interleaved: Wl[k/2*64 + col] = {W[k][col], W[k+1][col]}
// so each B fragment is a single aligned ds_load_b64 into an even VGPR pair
// (no repacking movs), and each A fragment is a single global_load_b64.
__global__ __launch_bounds__(THREADS) void lgcn_gemm(
    const float* __restrict__ A, const float* __restrict__ W,
    const float* __restrict__ bias, float* __restrict__ out,
    int n, int n_strips) {
  __shared__ v2f Wl[64 * 64];           // 32 KB
  for (int i = threadIdx.x; i < 64 * 64; i += THREADS) {
    int k2  = i >> 6;
    int col = i & 63;
    v2f p;
    p.x = W[(2 * k2) * 64 + col];
    p.y = W[(2 * k2 + 1) * 64 + col];
    Wl[i] = p;
  }
  __syncthreads();

  int strip = blockIdx.x * (THREADS / 32) + ((int)threadIdx.x >> 5);
  if (strip >= n_strips) return;        // uniform per-wave: EXEC stays all-1s
  int lane = (int)threadIdx.x & 31;
  int hl = lane >> 4;                   // 0: K pair {0,1}; 1: K pair {2,3}
  int l16 = lane & 15;

  int rowA = strip * 16 + l16;
  if (rowA >= n) rowA = n - 1;          // pad rows read row n-1, never stored
  const float* arow = A + (size_t)rowA * 128;

  v8f c0 = {}; v8f c1 = {}; v8f c2 = {}; v8f c3 = {};
#pragma unroll
  for (int k0 = 0; k0 < 128; k0 += 4) {
    // A 16x4: lanes 0-15 hold K={k0,k0+1}, lanes 16-31 hold K={k0+2,k0+3}
    v2f a = *(const v2f*)(arow + k0 + hl * 2);   // 8B-aligned -> global_load_b64
    const v2f* wrow = Wl + ((k0 >> 1) + hl) * 64; // pair-row ka/2
    v2f b0 = wrow[l16];
    v2f b1 = wrow[l16 + 16];
    v2f b2 = wrow[l16 + 32];
    v2f b3 = wrow[l16 + 48];
    c0 = __builtin_amdgcn_wmma_f32_16x16x4_f32(false, a, false, b0, (short)0, c0, false, false);
    c1 = __builtin_amdgcn_wmma_f32_16x16x4_f32(false, a, false, b1, (short)0, c1, false, false);
    c2 = __builtin_amdgcn_wmma_f32_16x16x4_f32(false, a, false, b2, (short)0, c2, false, false);
    c3 = __builtin_amdgcn_wmma_f32_16x16x4_f32(false, a, false, b3, (short)0, c3, false, false);
  }

  float bb0 = bias[l16], bb1 = bias[l16 + 16], bb2 = bias[l16 + 32], bb3 = bias[l16 + 48];
  const float s = 0.25f;                // /(NUM_LAYERS+1), folded post-GEMM (linear)
#pragma unroll
  for (int r = 0; r < 8; ++r) {
    int row = strip * 16 + hl * 8 + r;  // C/D: VGPR r -> M = r + 8*(lane/16)
    if (row < n) {
      float* orow = out + (size_t)row * 64;
      orow[l16]      = c0[r] * s + bb0;
      orow[l16 + 16] = c1[r] * s + bb1;
      orow[l16 + 32] = c2[r] * s + bb2;
      orow[l16 + 48] = c3[r] * s + bb3;
    }
  }
}

extern "C" void kernel_launch(void* const* d_in, const int* in_sizes, int n_in,
                              void* d_out, int out_size, void* d_ws, size_t ws_size,
                              hipStream_t stream) {
  const float*     x    = (const float*)d_in[0];
  const float*     W    = (const float*)d_in[1];
  const float*     b    = (const float*)d_in[2];
  const long long* edge = (const long long*)d_in[3];
  float* out = (float*)d_out;

  const int n128 = in_sizes[0];       // N * 128
  const int n    = n128 / 128;        // 100000
  const int E    = in_sizes[3] / 2;   // 625000

  // Workspace carve-out (256B aligned): deg/dis | norm | total | hA | hB
  char* ws = (char*)d_ws;
  size_t off = 0;
  auto carve = [&](size_t bytes) -> void* {
    void* p = ws + off;
    off = (off + bytes + 255) & ~(size_t)255;
    return p;
  };
  float* deg   = (float*)carve((size_t)n * 4);          // becomes dis in place
  float* normv = (float*)carve((size_t)E * 4);
  float* total = (float*)carve((size_t)n128 * 4);
  float* hA    = (float*)carve((size_t)n128 * 4);
  float* hB    = (float*)carve((size_t)n128 * 4);
  (void)ws_size; (void)n_in; (void)out_size;

  const unsigned gN128 = (unsigned)((n128 + THREADS - 1) / THREADS);
  const unsigned gN    = (unsigned)((n + THREADS - 1) / THREADS);
  const unsigned gE    = (unsigned)((E + THREADS - 1) / THREADS);
  const unsigned gScat = (unsigned)(((size_t)E * 32 + THREADS - 1) / THREADS);

  lgcn_init<<<gN128, THREADS, 0, stream>>>(x, total, hA, deg, n128, n);
  lgcn_deg <<<gE,    THREADS, 0, stream>>>(edge, deg, E);
  lgcn_dis <<<gN,    THREADS, 0, stream>>>(deg, n);
  lgcn_norm<<<gE,    THREADS, 0, stream>>>(edge, deg, normv, E);

  // Layer 1: x -> hA (hA zeroed by init)
  lgcn_scatter<<<gScat, THREADS, 0, stream>>>(edge, normv, x,  hA, E);
  lgcn_accum  <<<gN128, THREADS, 0, stream>>>(total, hA, hB, n128);
  // Layer 2: hA -> hB (hB zeroed by previous accum)
  lgcn_scatter<<<gScat, THREADS, 0, stream>>>(edge, normv, hA, hB, E);
  lgcn_accum  <<<gN128, THREADS, 0, stream>>>(total, hB, hA, n128);
  // Layer 3: hB -> hA (hA zeroed by previous accum)
  lgcn_scatter<<<gScat, THREADS, 0, stream>>>(edge, normv, hB, hA, E);
  lgcn_accum  <<<gN128, THREADS, 0, stream>>>(total, hA, hB, n128);

  const int strips = (n + 15) / 16;
  const unsigned gGemm = (unsigned)((strips + (THREADS / 32) - 1) / (THREADS / 32));
  lgcn_gemm<<<gGemm, THREADS, 0, stream>>>(total, W, b, out, n, strips);
}